// Double_qlearning_31782757990809
// MI455X (gfx1250) — compile-verified
//
#include <hip/hip_runtime.h>
#include <hip/hip_bf16.h>
#include <stdint.h>

// D = A(16x4) * B(4x16) + C(16x16), f32 throughout.
typedef __attribute__((ext_vector_type(2))) float v2f;
typedef __attribute__((ext_vector_type(4))) float v4f;
typedef __attribute__((ext_vector_type(8))) float v8f;

#define BATCH          8192
#define ITEMS          10000
#define ROWS_PER_BLOCK 16
#define KC             128               // K-tile width (floats)
#define LDS_STRIDE     (KC + 4)          // 132: 4-bank skew per row -> conflict-free A-fragment reads

__global__ void dqn_zero_out_kernel(float* __restrict__ out, int n) {
  int i = blockIdx.x * blockDim.x + threadIdx.x;
  if (i < n) out[i] = 0.0f;
}

__global__ __launch_bounds__(256)
void dqn_loss_wmma_kernel(const float* __restrict__ hs,      // [B, N]
                          const float* __restrict__ tq,      // [B, N]
                          const float* __restrict__ rew,     // [B]
                          const float* __restrict__ disc_p,  // [1]
                          const uint8_t* __restrict__ done,  // [B] (bool8)
                          const float* __restrict__ W,       // [1, N]
                          const float* __restrict__ b_p,     // [1]
                          float* __restrict__ out) {         // [1]
  __shared__ float lds_q[ROWS_PER_BLOCK * LDS_STRIDE];
  __shared__ float lds_t[ROWS_PER_BLOCK * LDS_STRIDE];
  __shared__ float lds_w[KC];
  __shared__ float red_q[8 * 16];
  __shared__ float red_t[8 * 16];
  __shared__ float finals[16];

  const int tid  = threadIdx.x;
  const int lane = tid & 31;           // wave32
  const int wave = tid >> 5;           // 8 waves / block
  const int row0 = blockIdx.x * ROWS_PER_BLOCK;

  // A-fragment layout (f32 16x4): lanes 0-15 = rows, K={0,1}; lanes 16-31 = rows, K={2,3}.
  const int sel  = (lane >= 16) ? 2 : 0;
  const int arow = lane & 15;

  v8f acc_q = {};   // 16x16 f32 accumulator (all columns identical: B is W broadcast)
  v8f acc_t = {};

  for (int k0 = 0; k0 < ITEMS; k0 += KC) {
    __syncthreads();   // previous tile fully consumed

    // ---- cooperative coalesced tile load: 16 rows x 128 floats per matrix ----
    // 512 float4 per matrix / 256 threads = 2 each; zero-fill past N (10000 % 4 == 0).
    #pragma unroll
    for (int i = tid; i < ROWS_PER_BLOCK * (KC / 4); i += 256) {
      const int r  = i >> 5;       // 32 float4 per row
      const int c4 = i & 31;
      const int k  = k0 + c4 * 4;
      v4f vq = {}, vt = {};
      if (k + 4 <= ITEMS) {
        vq = *(const v4f*)(hs + (size_t)(row0 + r) * ITEMS + k);
        vt = *(const v4f*)(tq + (size_t)(row0 + r) * ITEMS + k);
      }
      *(v4f*)(&lds_q[r * LDS_STRIDE + c4 * 4]) = vq;
      *(v4f*)(&lds_t[r * LDS_STRIDE + c4 * 4]) = vt;
    }
    if (tid < KC / 4) {            // stage W chunk (broadcast B operand source)
      const int k = k0 + tid * 4;
      v4f vw = {};
      if (k + 4 <= ITEMS) vw = *(const v4f*)(W + k);
      *(v4f*)(&lds_w[tid * 4]) = vw;
    }
    __syncthreads();

    // ---- each wave consumes its private K=16 slice: 4 WMMA k-steps of K=4 ----
    #pragma unroll
    for (int s = 0; s < 4; ++s) {
      const int kk = wave * 16 + s * 4 + sel;                  // even -> 8B-aligned ds reads
      const v2f a_q = *(const v2f*)(&lds_q[arow * LDS_STRIDE + kk]);
      const v2f a_t = *(const v2f*)(&lds_t[arow * LDS_STRIDE + kk]);
      const v2f b_w = *(const v2f*)(&lds_w[kk]);               // W broadcast to all 16 columns
      acc_q = __builtin_amdgcn_wmma_f32_16x16x4_f32(
          /*neg_a=*/false, a_q, /*neg_b=*/false, b_w,
          /*c_mod=*/(short)0, acc_q, /*reuse_a=*/false, /*reuse_b=*/false);
      acc_t = __builtin_amdgcn_wmma_f32_16x16x4_f32(
          false, a_t, false, b_w, (short)0, acc_t, false, false);
    }
  }

  // ---- combine 8 wave partials. D layout: lane n holds column n; VGPR j:
  // lanes 0-15 -> row j, lanes 16-31 -> row 8+j. Columns identical (B broadcast),
  // so lane 0 carries rows 0-7, lane 16 carries rows 8-15.
  if (lane == 0) {
    #pragma unroll
    for (int j = 0; j < 8; ++j) {
      red_q[wave * 16 + j] = acc_q[j];
      red_t[wave * 16 + j] = acc_t[j];
    }
  }
  if (lane == 16) {
    #pragma unroll
    for (int j = 0; j < 8; ++j) {
      red_q[wave * 16 + 8 + j] = acc_q[j];
      red_t[wave * 16 + 8 + j] = acc_t[j];
    }
  }
  __syncthreads();

  // ---- epilogue: relu(fc(.)), TD error, masked |.| ----
  if (tid < 16) {
    float q = 0.0f, nq = 0.0f;
    #pragma unroll
    for (int w = 0; w < 8; ++w) {
      q  += red_q[w * 16 + tid];
      nq += red_t[w * 16 + tid];
    }
    const float bias = b_p[0];
    const float disc = disc_p[0];
    q  = fmaxf(q + bias, 0.0f);
    nq = fmaxf(nq + bias, 0.0f);
    const int grow = row0 + tid;
    float v = fabsf(rew[grow] + disc * nq - q);
    if (done[grow]) v = 0.0f;
    finals[tid] = v;
  }
  __syncthreads();

  if (tid == 0) {
    float s = 0.0f;
    #pragma unroll
    for (int j = 0; j < 16; ++j) s += finals[j];
    atomicAdd(out, s * (1.0f / (float)BATCH));   // pre-scaled mean
  }
}

extern "C" void kernel_launch(void* const* d_in, const int* in_sizes, int n_in,
                              void* d_out, int out_size, void* d_ws, size_t ws_size,
                              hipStream_t stream) {
  // setup_inputs() order:
  // 0 hidden_states [B*N] f32, 1 actions [B] i32 (unused), 2 rewards [B] f32,
  // 3 discount [1] f32, 4 targetQs_s [B*N] f32, 5 is_done [B] bool8,
  // 6 W [1*N] f32, 7 b [1] f32
  const float*   hs   = (const float*)d_in[0];
  const float*   rew  = (const float*)d_in[2];
  const float*   disc = (const float*)d_in[3];
  const float*   tq   = (const float*)d_in[4];
  const uint8_t* done = (const uint8_t*)d_in[5];
  const float*   W    = (const float*)d_in[6];
  const float*   b    = (const float*)d_in[7];
  float*         out  = (float*)d_out;

  // out is accumulated via atomics; re-zero every call (harness poisons once, never re-poisons).
  dqn_zero_out_kernel<<<1, 64, 0, stream>>>(out, out_size);

  const int blocks = BATCH / ROWS_PER_BLOCK;   // 512
  dqn_loss_wmma_kernel<<<blocks, 256, 0, stream>>>(hs, tq, rew, disc, done, W, b, out);
}